// GNN_Encoder_21320217657349
// MI455X (gfx1250) — compile-verified
//
#include <hip/hip_runtime.h>

// ---------------------------------------------------------------------------
// GCN layer for MI455X (gfx1250, wave32):
//   support = x @ W          (fp32 WMMA 16x16x4)
//   out     = relu(segment_sum(support[src] * ew, dst) + b)
// ---------------------------------------------------------------------------

typedef __attribute__((ext_vector_type(2))) float v2f;
typedef __attribute__((ext_vector_type(8))) float v8f;

#define N_NODES 100000
#define NFEAT   256
#define NHID    128
#define ROW_TILES (N_NODES / 16)       // 6250 row tiles of 16
#define WAVES_PER_BLOCK 2              // 6250 % 2 == 0 -> no tail guards
#define GEMM_BLOCK (WAVES_PER_BLOCK * 32)
#define KPANEL 64                      // K-depth of W panel staged in LDS
#define SW_STRIDE 144                  // padded row stride (banks disjoint for
                                       // lane halves reading rows k and k+2)

// ---------------------------------------------------------------------------
// Kernel 1: zero the accumulator (d_out) — must run every launch (atomics).
// ---------------------------------------------------------------------------
__global__ void zero_out_kernel(float4* __restrict__ out4, int n4) {
    int i = blockIdx.x * blockDim.x + threadIdx.x;
    if (i < n4) out4[i] = float4{0.f, 0.f, 0.f, 0.f};
}

// ---------------------------------------------------------------------------
// Kernel 2: support = x @ W via V_WMMA_F32_16X16X4_F32.
// One wave computes a 16x128 output slab (8 accumulator tiles of 16x16),
// iterating K in steps of 4. W K-panels staged through padded LDS.
// A-frag layout (f32 16x4): lanes 0-15 -> rows, K={k,k+1}; lanes 16-31 -> K={k+2,k+3}.
// B-frag layout (f32 4x16): lanes 0-15 -> cols, K={k,k+1}; lanes 16-31 -> K={k+2,k+3}.
// ---------------------------------------------------------------------------
__global__ __launch_bounds__(GEMM_BLOCK)
void gemm_xw_wmma(const float* __restrict__ x, const float* __restrict__ W,
                  float* __restrict__ support) {
    __shared__ float sW[KPANEL * SW_STRIDE];   // 64 * 144 * 4B = 36 KB

    const int tid   = threadIdx.x;
    const int wave  = tid >> 5;
    const int lane  = tid & 31;
    const int rt    = blockIdx.x * WAVES_PER_BLOCK + wave;  // row tile < 6250 always
    const int col0  = lane & 15;
    const int khalf = (lane >> 4) << 1;        // 0 for lanes 0-15, 2 for 16-31
    const int row   = rt * 16 + col0;          // A-matrix row owned by this lane

    v8f acc[8];
#pragma unroll
    for (int t = 0; t < 8; ++t) acc[t] = v8f{};

    for (int ks = 0; ks < NFEAT; ks += KPANEL) {
        __syncthreads();
        // Cooperatively stage a 64x128 panel of W into LDS (row stride 144).
        // 8192 floats = 2048 float4; 64 threads x 32 float4 each.
        {
            const float4* gW4 = (const float4*)(W + (size_t)ks * NHID);
            float4*       sW4 = (float4*)sW;   // dst row stride = 36 float4
#pragma unroll
            for (int j = 0; j < 2048 / GEMM_BLOCK; ++j) {
                const int c    = tid + GEMM_BLOCK * j;   // global float4 index
                const int krow = c >> 5;                 // 32 float4 per W row
                const int colc = c & 31;
                sW4[krow * (SW_STRIDE / 4) + colc] = gW4[c];
            }
        }
        __syncthreads();

#pragma unroll
        for (int k4 = 0; k4 < KPANEL; k4 += 4) {
            const int kk = k4 + khalf;
            // A fragment: 2 consecutive fp32 along K (8B aligned: kk is even).
            v2f a;
            {
                const float2 av =
                    *(const float2*)(x + (size_t)row * NFEAT + ks + kk);
                a.x = av.x;
                a.y = av.y;
            }
#pragma unroll
            for (int t = 0; t < 8; ++t) {
                const int col = t * 16 + col0;
                v2f bf;
                bf.x = sW[kk * SW_STRIDE + col];
                bf.y = sW[(kk + 1) * SW_STRIDE + col];
                acc[t] = __builtin_amdgcn_wmma_f32_16x16x4_f32(
                    /*neg_a=*/false, a, /*neg_b=*/false, bf,
                    /*c_mod=*/(short)0, acc[t],
                    /*reuse_a=*/false, /*reuse_b=*/false);
            }
        }
    }

    // C/D layout: VGPR r holds M = r (lanes 0-15) and M = 8+r (lanes 16-31).
    const int m0 = rt * 16 + ((lane >> 4) << 3);
#pragma unroll
    for (int t = 0; t < 8; ++t) {
        const int col = t * 16 + col0;
#pragma unroll
        for (int r = 0; r < 8; ++r)
            support[(size_t)(m0 + r) * NHID + col] = acc[t][r];
    }
}

// ---------------------------------------------------------------------------
// Kernel 3: edge gather-scale-scatter. One wave per edge; each lane handles
// 4 consecutive features (float4 gather from L2-resident support, then 4
// global_atomic_add_f32 into the L2-resident accumulator).
// ---------------------------------------------------------------------------
__global__ __launch_bounds__(256)
void edge_scatter(const float* __restrict__ support,
                  const int* __restrict__ esrc, const int* __restrict__ edst,
                  const float* __restrict__ ew, float* __restrict__ out,
                  int n_edges) {
    const int e = blockIdx.x * 8 + (threadIdx.x >> 5);
    if (e >= n_edges) return;
    const int lane = threadIdx.x & 31;

    const int   s = esrc[e];
    const int   d = edst[e];
    const float w = ew[e];

    const float4 v = ((const float4*)support)[(size_t)s * (NHID / 4) + lane];
    float* o = out + (size_t)d * NHID + lane * 4;
    atomicAdd(o + 0, v.x * w);
    atomicAdd(o + 1, v.y * w);
    atomicAdd(o + 2, v.z * w);
    atomicAdd(o + 3, v.w * w);
}

// ---------------------------------------------------------------------------
// Kernel 4: out = relu(out + b), float4-vectorized. b has 128 floats = 32 float4.
// ---------------------------------------------------------------------------
__global__ void bias_relu(float4* __restrict__ out4,
                          const float4* __restrict__ b4, int n4) {
    int i = blockIdx.x * blockDim.x + threadIdx.x;
    if (i < n4) {
        float4 v  = out4[i];
        float4 bb = b4[i & 31];
        v.x = fmaxf(v.x + bb.x, 0.f);
        v.y = fmaxf(v.y + bb.y, 0.f);
        v.z = fmaxf(v.z + bb.z, 0.f);
        v.w = fmaxf(v.w + bb.w, 0.f);
        out4[i] = v;
    }
}

// ---------------------------------------------------------------------------
// Launch: inputs in setup_inputs() order: x, edge_src, edge_dst, edge_weight, W, b
// ---------------------------------------------------------------------------
extern "C" void kernel_launch(void* const* d_in, const int* in_sizes, int n_in,
                              void* d_out, int out_size, void* d_ws,
                              size_t ws_size, hipStream_t stream) {
    const float* x    = (const float*)d_in[0];
    const int*   esrc = (const int*)d_in[1];
    const int*   edst = (const int*)d_in[2];
    const float* ew   = (const float*)d_in[3];
    const float* W    = (const float*)d_in[4];
    const float* b    = (const float*)d_in[5];
    float*       out  = (float*)d_out;
    float*       supp = (float*)d_ws;          // 100000*128*4B = 51.2 MB scratch

    const int n_edges = in_sizes[1];
    const int n4      = N_NODES * NHID / 4;    // 3.2M float4

    zero_out_kernel<<<(n4 + 255) / 256, 256, 0, stream>>>((float4*)out, n4);

    gemm_xw_wmma<<<ROW_TILES / WAVES_PER_BLOCK, GEMM_BLOCK, 0, stream>>>(
        x, W, supp);

    edge_scatter<<<(n_edges + 7) / 8, 256, 0, stream>>>(supp, esrc, edst, ew,
                                                        out, n_edges);

    bias_relu<<<(n4 + 255) / 256, 256, 0, stream>>>((float4*)out,
                                                    (const float4*)b, n4);
}